// Joint_11725260718246
// MI455X (gfx1250) — compile-verified
//
#include <hip/hip_runtime.h>
#include <hip/hip_bf16.h>
#include <math.h>

// ---------------- problem constants ----------------
#define BN    16
#define EMB   128
#define H0    256
#define UPD   512
#define NPIX  (UPD*UPD)        // 262144
#define RAD   60
#define PS    (2*RAD)          // 120
#define DOT2  100              // DOT=10 -> DOT^2
#define COEF  1.5f

typedef __attribute__((ext_vector_type(2))) float v2f;
typedef __attribute__((ext_vector_type(8))) float v8f;

// ---------------- helpers ----------------
__device__ __forceinline__ float gat(const float* img, float yf, float xf, int H, int W) {
  bool valid = (xf >= 0.f) && (xf <= (float)(W - 1)) && (yf >= 0.f) && (yf <= (float)(H - 1));
  int yi = (int)fminf(fmaxf(yf, 0.f), (float)(H - 1));
  int xi = (int)fminf(fmaxf(xf, 0.f), (float)(W - 1));
  float v = img[yi * W + xi];
  return valid ? v : 0.f;
}

__device__ __forceinline__ float gsample(const float* img, float gx, float gy, int H, int W) {
  float x = ((gx + 1.f) * (float)W - 1.f) * 0.5f;
  float y = ((gy + 1.f) * (float)H - 1.f) * 0.5f;
  float x0f = floorf(x), y0f = floorf(y);
  float wx = x - x0f, wy = y - y0f;
  float v00 = gat(img, y0f,       x0f,       H, W);
  float v01 = gat(img, y0f,       x0f + 1.f, H, W);
  float v10 = gat(img, y0f + 1.f, x0f,       H, W);
  float v11 = gat(img, y0f + 1.f, x0f + 1.f, H, W);
  return v00*(1.f-wy)*(1.f-wx) + v01*(1.f-wy)*wx + v10*wy*(1.f-wx) + v11*wy*wx;
}

__device__ __forceinline__ void inv_affine(const float* A, float* o) {
  // A = [a b c; d e f; 0 0 1]; o = top-2 rows of inverse
  float a=A[0], b=A[1], c=A[2], d=A[3], e=A[4], f=A[5];
  float id = 1.f / (a*e - b*d);
  o[0] =  e*id; o[1] = -b*id; o[2] = (b*f - c*e)*id;
  o[3] = -d*id; o[4] =  a*id; o[5] = (c*d - a*f)*id;
}

// ---------------- kernels ----------------
__global__ void k_copyf(const float* __restrict__ s, float* __restrict__ d, int n) {
  int i = blockIdx.x * blockDim.x + threadIdx.x;
  if (i < n) d[i] = s[i];
}

__global__ void k_zero(float* p, int n) {
  int i = blockIdx.x * blockDim.x + threadIdx.x;
  if (i < n) p[i] = 0.f;
}

__global__ void k_invert(const float* __restrict__ ss, const float* __restrict__ rot,
                         float* __restrict__ inv1, float* __restrict__ inv2) {
  int b = threadIdx.x;
  if (b < BN) {
    inv_affine(ss  + b*6, inv1 + b*6);
    inv_affine(rot + b*6, inv2 + b*6);
  }
}

// base = relu(k_out @ W_dec): (16x128)@(128x65536) via V_WMMA_F32_16X16X4_F32.
// One wave per 16-wide N tile; K accumulated over 32 chained WMMAs.
__global__ void k_gemm_wmma(const float* __restrict__ kout, const float* __restrict__ W,
                            float* __restrict__ base) {
  int wave = (blockIdx.x * blockDim.x + threadIdx.x) >> 5;   // 0..4095
  int lane = threadIdx.x & 31;
  int half = lane >> 4;          // 0: K pair {k,k+1}, 1: {k+2,k+3}
  int l16  = lane & 15;
  int n0   = wave << 4;          // N tile base

  v8f acc = {};
#pragma unroll 4
  for (int k = 0; k < EMB; k += 4) {
    int kb = k + 2*half;
    // A fragment (16x4 f32): lane l16 holds row M=l16, K = kb, kb+1
    v2f a;
    a.x = kout[l16 * EMB + kb];
    a.y = kout[l16 * EMB + kb + 1];
    // B fragment (4x16 f32): lane l16 holds col N=n0+l16, K = kb, kb+1
    v2f bf;
    bf.x = W[(size_t)kb       * (H0*H0) + n0 + l16];
    bf.y = W[(size_t)(kb + 1) * (H0*H0) + n0 + l16];
    acc = __builtin_amdgcn_wmma_f32_16x16x4_f32(
        /*neg_a=*/false, a, /*neg_b=*/false, bf,
        /*c_mod=*/(short)0, acc, /*reuse_a=*/false, /*reuse_b=*/false);
  }
  // D layout: VGPR r, lane -> M = r + 8*half, N = n0 + l16 ; ReLU epilogue
#pragma unroll
  for (int r = 0; r < 8; ++r) {
    float v = acc[r];
    v = v > 0.f ? v : 0.f;
    base[(size_t)(r + 8*half) * (H0*H0) + n0 + l16] = v;
  }
}

// bilinear resize 256x256 -> 512x512 per batch
__global__ void k_resize(const float* __restrict__ base, float* __restrict__ out) {
  int idx = blockIdx.x * blockDim.x + threadIdx.x;          // 16*512*512
  int b = idx >> 18; int p = idx & (NPIX - 1);
  int i = p >> 9, j = p & (UPD - 1);
  float ys = fminf(fmaxf((i + 0.5f) * 0.5f - 0.5f, 0.f), 255.f);
  float xs = fminf(fmaxf((j + 0.5f) * 0.5f - 0.5f, 0.f), 255.f);
  int y0 = (int)floorf(ys), x0 = (int)floorf(xs);
  int y1 = min(y0 + 1, 255), x1 = min(x0 + 1, 255);
  float wy = ys - (float)y0, wx = xs - (float)x0;
  const float* im = base + (size_t)b * (H0*H0);
  float Ia = im[y0*H0 + x0], Ib = im[y0*H0 + x1];
  float Ic = im[y1*H0 + x0], Id = im[y1*H0 + x1];
  out[idx] = Ia*(1.f-wy)*(1.f-wx) + Ib*(1.f-wy)*wx + Ic*wy*(1.f-wx) + Id*wy*wx;
}

// dst[b] = grid_sample(src[b], affine_grid(theta[b], 512,512))
__global__ void k_warp(const float* __restrict__ src, const float* __restrict__ theta,
                       float* __restrict__ dst) {
  int idx = blockIdx.x * blockDim.x + threadIdx.x;
  int b = idx >> 18; int p = idx & (NPIX - 1);
  int i = p >> 9, j = p & (UPD - 1);
  const float* t = theta + b*6;
  float X = (j + 0.5f) * (2.f / UPD) - 1.f;
  float Y = (i + 0.5f) * (2.f / UPD) - 1.f;
  float gx = t[0]*X + t[1]*Y + t[2];
  float gy = t[3]*X + t[4]*Y + t[5];
  dst[idx] = gsample(src + (size_t)b * NPIX, gx, gy, UPD, UPD);
}

// inner sample of a (un-batched) mask through grid2 at integer pixel (qy,qx)
__device__ __forceinline__ float sample_mask_g2(const float* mask, const float* t2, int qy, int qx) {
  float X = (qx + 0.5f) * (2.f / UPD) - 1.f;
  float Y = (qy + 0.5f) * (2.f / UPD) - 1.f;
  float gx = t2[0]*X + t2[1]*Y + t2[2];
  float gy = t2[3]*X + t2[4]*Y + t2[5];
  return gsample(mask, gx, gy, UPD, UPD);
}

// Pass A: rot_mask computed on the fly (double warp), store m; reduce sum(m), sum(pred_in*m)
__global__ void k_mask_sums(const float* __restrict__ masks, int mj,
                            const float* __restrict__ inv1, const float* __restrict__ inv2,
                            const float* __restrict__ pred_in,
                            float* __restrict__ m_out, float* __restrict__ red) {
  int b = blockIdx.x >> 10;                 // 1024 blocks per batch
  int p = ((blockIdx.x & 1023) << 8) + threadIdx.x;
  int i = p >> 9, j = p & (UPD - 1);
  const float* t1 = inv1 + b*6;
  const float* t2 = inv2 + b*6;
  const float* mask = masks + (size_t)mj * NPIX;

  // outer sample through grid1
  float X = (j + 0.5f) * (2.f / UPD) - 1.f;
  float Y = (i + 0.5f) * (2.f / UPD) - 1.f;
  float gx = t1[0]*X + t1[1]*Y + t1[2];
  float gy = t1[3]*X + t1[4]*Y + t1[5];
  float x = ((gx + 1.f) * UPD - 1.f) * 0.5f;
  float y = ((gy + 1.f) * UPD - 1.f) * 0.5f;
  float x0f = floorf(x), y0f = floorf(y);
  float wx = x - x0f, wy = y - y0f;
  float w4[4] = { (1.f-wy)*(1.f-wx), (1.f-wy)*wx, wy*(1.f-wx), wy*wx };
  float acc = 0.f;
#pragma unroll
  for (int d = 0; d < 4; ++d) {
    float yf = y0f + (float)(d >> 1);
    float xf = x0f + (float)(d & 1);
    if (xf >= 0.f && xf <= (float)(UPD-1) && yf >= 0.f && yf <= (float)(UPD-1))
      acc += w4[d] * sample_mask_g2(mask, t2, (int)yf, (int)xf);
  }
  float m = (acc >= 0.5f) ? 1.f : 0.f;
  size_t off = (size_t)b * NPIX + p;
  m_out[off] = m;
  float ni = pred_in[off] * m;

  __shared__ float s0[256], s1[256];
  int tid = threadIdx.x;
  s0[tid] = m; s1[tid] = ni;
  __syncthreads();
  for (int s = 128; s > 0; s >>= 1) {
    if (tid < s) { s0[tid] += s0[tid+s]; s1[tid] += s1[tid+s]; }
    __syncthreads();
  }
  if (tid == 0) {
    atomicAdd(&red[b],      s0[0]);   // sum(m)
    atomicAdd(&red[BN + b], s1[0]);   // sum(new_img)
  }
}

// Pass B: selection + centroid sums
__global__ void k_sel_sums(const float* __restrict__ m_in, const float* __restrict__ pred_in,
                           float* __restrict__ red) {
  int b = blockIdx.x >> 10;
  int p = ((blockIdx.x & 1023) << 8) + threadIdx.x;
  int i = p >> 9, j = p & (UPD - 1);
  size_t off = (size_t)b * NPIX + p;
  float m = m_in[off];
  float ni = pred_in[off] * m;
  float mean = red[BN + b] / fmaxf(red[b], 1.f);
  bool sel = (m > 0.f) && (ni > COEF * mean);
  float w = sel ? ni : 0.f;

  __shared__ float s0[256], s1[256], s2[256];
  int tid = threadIdx.x;
  s0[tid] = w; s1[tid] = w * (float)i; s2[tid] = w * (float)j;
  __syncthreads();
  for (int s = 128; s > 0; s >>= 1) {
    if (tid < s) { s0[tid] += s0[tid+s]; s1[tid] += s1[tid+s]; s2[tid] += s2[tid+s]; }
    __syncthreads();
  }
  if (tid == 0) {
    atomicAdd(&red[2*BN + b], s0[0]);   // tot
    atomicAdd(&red[3*BN + b], s1[0]);   // sum(w*row)
    atomicAdd(&red[4*BN + b], s2[0]);   // sum(w*col)
  }
}

__global__ void k_centroid(const float* __restrict__ red, int* __restrict__ ctr) {
  int b = threadIdx.x;
  if (b < BN) {
    float tot = red[2*BN + b] + 1e-8f;
    float cx = red[3*BN + b] / tot;   // row centroid
    float cy = red[4*BN + b] / tot;   // col centroid
    ctr[2*b]     = (int)fminf(fmaxf(rintf(cx), (float)RAD), (float)(UPD - RAD));
    ctr[2*b + 1] = (int)fminf(fmaxf(rintf(cy), (float)RAD), (float)(UPD - RAD));
  }
}

// Per-batch patch revise: stage 120x120 patch in LDS, adj-divide in circle,
// resample through inv1, write back into img.
__global__ void k_patch(float* __restrict__ img, const float* __restrict__ inv1,
                        const float* __restrict__ adj_mask, const int* __restrict__ ctr) {
  __shared__ float patch[PS * PS];
  int b = blockIdx.x;
  int cxi = ctr[2*b], cyi = ctr[2*b + 1];
  float adj = adj_mask[b];
  float* im = img + (size_t)b * NPIX;

  for (int p = threadIdx.x; p < PS*PS; p += blockDim.x) {
    int pi = p / PS, pj = p % PS;
    float v = im[(cxi - RAD + pi) * UPD + (cyi - RAD + pj)];
    int dy = pi - RAD, dx = pj - RAD;
    if (dy*dy + dx*dx <= DOT2) v /= adj;
    patch[p] = v;
  }
  __syncthreads();

  const float* t = inv1 + b*6;
  for (int p = threadIdx.x; p < PS*PS; p += blockDim.x) {
    int pi = p / PS, pj = p % PS;
    float X = (pj + 0.5f) * (2.f / PS) - 1.f;
    float Y = (pi + 0.5f) * (2.f / PS) - 1.f;
    float gx = t[0]*X + t[1]*Y + t[2];
    float gy = t[3]*X + t[4]*Y + t[5];
    im[(cxi - RAD + pi) * UPD + (cyi - RAD + pj)] = gsample(patch, gx, gy, PS, PS);
  }
}

// ---------------- host launcher ----------------
extern "C" void kernel_launch(void* const* d_in, const int* in_sizes, int n_in,
                              void* d_out, int out_size, void* d_ws, size_t ws_size,
                              hipStream_t stream) {
  const float* x     = (const float*)d_in[0];   // 16*1*256*256
  const float* kout  = (const float*)d_in[1];   // 16*128
  const float* W_dec = (const float*)d_in[2];   // 128*65536
  const float* ss    = (const float*)d_in[3];   // 16*2*3
  const float* rot   = (const float*)d_in[4];   // 16*2*3
  const float* masks = (const float*)d_in[5];   // 4*512*512
  const float* adj   = (const float*)d_in[6];   // 16
  (void)in_sizes; (void)n_in; (void)out_size; (void)ws_size;

  float* out      = (float*)d_out;
  float* out_x    = out;                         // 1,048,576
  float* out_base = out + BN*H0*H0;              // 4,194,304
  float* out_img  = out_base + (size_t)BN*NPIX;  // 4,194,304

  float* wsf    = (float*)d_ws;
  float* w_inv1 = wsf;                    // 96
  float* w_inv2 = w_inv1 + 96;            // 96
  float* w_red  = w_inv2 + 96;            // 80
  int*   w_ctr  = (int*)(w_red + 80);     // 32 ints
  float* w_base = wsf + 1024;             // 16*65536   =  4 MB
  float* w_tmp  = w_base + (size_t)BN*H0*H0;   // 16*262144 = 16 MB (pred_rot, then masks)
  float* w_pin  = w_tmp + (size_t)BN*NPIX;     // 16*262144 = 16 MB (pred_in)

  // 1) out0 = x
  k_copyf<<<(BN*H0*H0)/256, 256, 0, stream>>>(x, out_x, BN*H0*H0);
  // 2) affine inverses
  k_invert<<<1, 32, 0, stream>>>(ss, rot, w_inv1, w_inv2);
  // 3) base = relu(k_out @ W_dec)  -- WMMA f32 16x16x4
  k_gemm_wmma<<<(H0*H0/16)/8, 256, 0, stream>>>(kout, W_dec, w_base);
  // 4) base_inp = resize(base, 512,512) -> out1
  k_resize<<<(BN*NPIX)/256, 256, 0, stream>>>(w_base, out_base);
  // 5) pred_rot = sample(base_inp, grid2); pred_in = sample(pred_rot, grid1)
  k_warp<<<(BN*NPIX)/256, 256, 0, stream>>>(out_base, w_inv2, w_tmp);
  k_warp<<<(BN*NPIX)/256, 256, 0, stream>>>(w_tmp, w_inv1, w_pin);
  // 6) img starts as pred_in -> out2
  k_copyf<<<(BN*NPIX)/256, 256, 0, stream>>>(w_pin, out_img, BN*NPIX);
  // 7) sequential revise over 4 masks
  for (int j = 0; j < 4; ++j) {
    k_zero<<<1, 128, 0, stream>>>(w_red, 80);
    k_mask_sums<<<(BN*NPIX)/256, 256, 0, stream>>>(masks, j, w_inv1, w_inv2, w_pin, w_tmp, w_red);
    k_sel_sums<<<(BN*NPIX)/256, 256, 0, stream>>>(w_tmp, w_pin, w_red);
    k_centroid<<<1, 32, 0, stream>>>(w_red, w_ctr);
    k_patch<<<BN, 256, 0, stream>>>(out_img, w_inv1, adj, w_ctr);
  }
}